// CollapseAwareEmbedding_20770461844088
// MI455X (gfx1250) — compile-verified
//
#include <hip/hip_runtime.h>

typedef __attribute__((ext_vector_type(2))) float v2f;
typedef __attribute__((ext_vector_type(4))) float v4f;
typedef __attribute__((ext_vector_type(8))) float v8f;

#define S_IN   21
#define S_DIM  256
#define B_SZ   32
#define L_TOT  256
#define POS_DIM 64

// ---------------------------------------------------------------------------
// posenc(p, i) = (i odd ? cos : sin)(p * 10000^{-i/64})
// ---------------------------------------------------------------------------
__device__ __forceinline__ float posenc_elem(int p, int i) {
  // 10000^{-i/64} = exp(-i * ln(10000)/64), ln(10000)/64 = 0.14391157
  float freq = __expf(-(float)i * 0.14391157f);
  float ang  = (float)p * freq;
  return (i & 1) ? __cosf(ang) : __sinf(ang);
}

// ---------------------------------------------------------------------------
// Kernel 1: posW[p][h] = (posenc(p,:) @ W_pos)[h] + b_pos[h],  p in [0,128)
// One wave per 16x16 tile, K=64 via 16 steps of v_wmma_f32_16x16x4_f32.
// ---------------------------------------------------------------------------
__global__ void posw_kernel(const float* __restrict__ W_pos,
                            const float* __restrict__ b_pos,
                            float* __restrict__ posW) {
  int wave = (blockIdx.x * blockDim.x + threadIdx.x) >> 5;  // 0..127
  int lane = threadIdx.x & 31;
  int half = lane >> 4;
  int sub  = lane & 15;
  int m0 = (wave >> 4) * 16;   // p tile
  int n0 = (wave & 15) * 16;   // h tile
  int p   = m0 + sub;          // A row for this lane
  int col = n0 + sub;          // B col / D col for this lane

  v8f acc = {};
  for (int kk = 0; kk < POS_DIM; kk += 4) {
    int ka = kk + half * 2;
    v2f a, b;
    a.x = posenc_elem(p, ka);
    a.y = posenc_elem(p, ka + 1);
    b.x = W_pos[ka * S_DIM + col];
    b.y = W_pos[(ka + 1) * S_DIM + col];
    acc = __builtin_amdgcn_wmma_f32_16x16x4_f32(false, a, false, b,
                                                (short)0, acc, false, false);
  }
  float bp = b_pos[col];
  for (int r = 0; r < 8; ++r) {
    int row = m0 + r + half * 8;
    posW[row * S_DIM + col] = acc[r] + bp;
  }
}

// ---------------------------------------------------------------------------
// Kernel 2: s[b,l,:] . One wave per 16x16 output tile (8192 tiles).
// Global row = b*256 + l ; l==0 -> collapse row, else region GEMM + pos table.
// K = 21 padded to 24 -> 6 WMMA steps.
// ---------------------------------------------------------------------------
__global__ void s_kernel(const float* __restrict__ cdr3,
                         const float* __restrict__ pep,
                         const float* __restrict__ mhc,
                         const float* __restrict__ W_seq,
                         const float* __restrict__ b_seq,
                         const float* __restrict__ posW,
                         const float* __restrict__ collapse_token,
                         const float* __restrict__ collapse_weight,
                         const float* __restrict__ w_cdr3,
                         const float* __restrict__ w_pep,
                         const float* __restrict__ w_mhc,
                         float* __restrict__ s_out) {
  int wave = (blockIdx.x * blockDim.x + threadIdx.x) >> 5;  // 0..8191
  int lane = threadIdx.x & 31;
  int half = lane >> 4;
  int sub  = lane & 15;
  int m0 = (wave >> 4) * 16;   // global row tile (never spans batches: 16|256)
  int n0 = (wave & 15) * 16;

  // Per-lane A-row source pointer (row = m0+sub, shared by both lane halves)
  int arow = m0 + sub;
  int al = arow & 255;
  int ab = arow >> 8;
  const float* arp = nullptr;
  if (al >= 1 && al < 128)        arp = cdr3 + ((size_t)ab * 127 + (al - 1))  * S_IN;
  else if (al >= 128 && al < 192) arp = pep  + ((size_t)ab * 64  + (al - 128)) * S_IN;
  else if (al >= 192)             arp = mhc  + ((size_t)ab * 64  + (al - 192)) * S_IN;

  int col = n0 + sub;
  v8f acc = {};
  for (int kk = 0; kk < 24; kk += 4) {
    int ka = kk + half * 2;
    v2f a, b;
    a.x = (arp && ka     < S_IN) ? arp[ka]     : 0.0f;
    a.y = (arp && ka + 1 < S_IN) ? arp[ka + 1] : 0.0f;
    b.x = (ka     < S_IN) ? W_seq[ka * S_DIM + col]       : 0.0f;
    b.y = (ka + 1 < S_IN) ? W_seq[(ka + 1) * S_DIM + col] : 0.0f;
    acc = __builtin_amdgcn_wmma_f32_16x16x4_f32(false, a, false, b,
                                                (short)0, acc, false, false);
  }

  float bsc = b_seq[col];
  float wc0 = w_cdr3[0], wc1 = w_cdr3[1];
  float wp0 = w_pep[0],  wp1 = w_pep[1];
  float wm0 = w_mhc[0],  wm1 = w_mhc[1];
  float cw  = collapse_weight[0];
  float ctk = collapse_token[col];

  for (int r = 0; r < 8; ++r) {
    int grow = m0 + r + half * 8;       // = b*256 + l
    int l = grow & 255;
    float out;
    if (l == 0) {
      out = cw * ctk;
    } else {
      float w0, w1; int p;
      if (l < 128)      { w0 = wc0; w1 = wc1; p = l - 1;   }
      else if (l < 192) { w0 = wp0; w1 = wp1; p = l - 128; }
      else              { w0 = wm0; w1 = wm1; p = l - 192; }
      out = w0 * (acc[r] + bsc) + w1 * posW[p * S_DIM + col];
    }
    s_out[(size_t)grow * S_DIM + col] = out;
  }
}

// ---------------------------------------------------------------------------
// Kernel 3: z[b,i,j,k] — store-bandwidth bound (512 MB). 8-row table in LDS,
// pid(i,j) recomputed inline, non-temporal 128-bit coalesced stores.
// ---------------------------------------------------------------------------
__global__ void z_kernel(const float* __restrict__ W_p1,
                         const float* __restrict__ b_p1,
                         const float* __restrict__ W_p2,
                         const float* __restrict__ b_p2,
                         float* __restrict__ zout) {
  __shared__ __align__(16) float tab[8 * 64];
  for (int t = threadIdx.x; t < 512; t += blockDim.x) {
    int pid = t >> 6, k = t & 63;
    tab[t] = (k < 32) ? (W_p1[(pid >> 2) * 32 + k]        + b_p1[k])
                      : (W_p2[(pid & 3)  * 32 + (k - 32)] + b_p2[k - 32]);
  }
  __syncthreads();

  const long long total  = (long long)B_SZ * L_TOT * L_TOT * 16;  // float4 units
  const long long stride = (long long)gridDim.x * blockDim.x;
  for (long long q = (long long)blockIdx.x * blockDim.x + threadIdx.x;
       q < total; q += stride) {
    int k4 = (int)(q & 15);
    int j  = (int)((q >> 4)  & 255);
    int i  = (int)((q >> 12) & 255);
    // value independent of b = q >> 20
    int ri = (i == 0) ? 0 : (i < 128) ? 1 : (i < 192) ? 2 : 3;
    int rj = (j == 0) ? 0 : (j < 128) ? 1 : (j < 192) ? 2 : 3;
    int d  = i - j; if (d < 0) d = -d;
    int pid = 0;
    if (ri == 1 && rj == 1 && i != j)                 pid = 3;
    if (ri == 1 && rj == 1 && d == 1)                 pid = 2;
    if ((ri == 1 && rj >= 2) || (ri >= 2 && rj == 1)) pid = 4;
    if (ri == 2 && rj == 2)                           pid = 5;
    if (ri == 3 && rj == 3)                           pid = 6;
    if ((ri == 2 && rj == 3) || (ri == 3 && rj == 2)) pid = 7;
    if (ri == 0 || rj == 0)                           pid = 1;
    if (i == 0 && j == 0)                             pid = 0;

    v4f v = *(const v4f*)&tab[pid * 64 + k4 * 4];
    __builtin_nontemporal_store(v, (v4f*)zout + q);
  }
}

// ---------------------------------------------------------------------------
extern "C" void kernel_launch(void* const* d_in, const int* in_sizes, int n_in,
                              void* d_out, int out_size, void* d_ws, size_t ws_size,
                              hipStream_t stream) {
  const float* cdr3   = (const float*)d_in[0];
  const float* pep    = (const float*)d_in[1];
  const float* mhc    = (const float*)d_in[2];
  const float* W_seq  = (const float*)d_in[3];
  const float* b_seq  = (const float*)d_in[4];
  const float* W_pos  = (const float*)d_in[5];
  const float* b_pos  = (const float*)d_in[6];
  const float* W_p1   = (const float*)d_in[7];
  const float* b_p1   = (const float*)d_in[8];
  const float* W_p2   = (const float*)d_in[9];
  const float* b_p2   = (const float*)d_in[10];
  const float* ctok   = (const float*)d_in[11];
  const float* cw     = (const float*)d_in[12];
  const float* w_cdr3 = (const float*)d_in[13];
  const float* w_pep  = (const float*)d_in[14];
  const float* w_mhc  = (const float*)d_in[15];
  // d_in[16..18] = masks, unused by the reference computation

  float* s_out = (float*)d_out;                                 // 32*256*256
  float* z_out = s_out + (size_t)B_SZ * L_TOT * S_DIM;          // 32*256*256*64
  float* posW  = (float*)d_ws;                                  // 128*256 f32

  // 1) pos-emb table (WMMA f32 16x16x4): 128 waves = 16 blocks x 8 waves
  posw_kernel<<<16, 256, 0, stream>>>(W_pos, b_pos, posW);
  // 2) s (WMMA): 8192 tiles / 8 waves per block = 1024 blocks
  s_kernel<<<1024, 256, 0, stream>>>(cdr3, pep, mhc, W_seq, b_seq, posW,
                                     ctok, cw, w_cdr3, w_pep, w_mhc, s_out);
  // 3) z streaming NT stores: grid-stride, 16384 blocks x 256 threads
  z_kernel<<<16384, 256, 0, stream>>>(W_p1, b_p1, W_p2, b_p2, z_out);
}